// GConvLSTM_Temporal_35605278884398
// MI455X (gfx1250) — compile-verified
//
#include <hip/hip_runtime.h>
#include <hip/hip_bf16.h>

// GConvLSTM (Cheb K=2) forward for MI455X / gfx1250.
// - sparse normalized-Laplacian scatter: wave-per-edge, coalesced global_atomic_add_f32
//   (memory/atomic bound; whole working set ~100MB is L2-resident at 192MB L2)
// - dense gate GEMMs: V_WMMA_F32_16X16X4_F32 (exact fp32, wave32 WMMA),
//   A-tile staged to LDS with GLOBAL_LOAD_ASYNC_TO_LDS_B128 (ASYNCcnt path)

#define N_NODES 50000
#define N_EDGES 1600000
#define HIDDEN  32
#define HORIZON 12

typedef __attribute__((ext_vector_type(2))) float v2f;
typedef __attribute__((ext_vector_type(8))) float v8f;

__device__ __forceinline__ float sigmoidf_(float x) {
  return 1.0f / (1.0f + __expf(-x));
}

// ---------------- degree / dinv ----------------
__global__ void k_zero_deg(float* __restrict__ deg) {
  int i = blockIdx.x * 256 + threadIdx.x;
  if (i < N_NODES) deg[i] = 0.0f;
}

__global__ void k_deg_accum(const long long* __restrict__ ei,
                            const float* __restrict__ w,
                            float* __restrict__ deg) {
  int e = blockIdx.x * 256 + threadIdx.x;
  if (e < N_EDGES) atomicAdd(&deg[(int)ei[e]], w[e]);   // ei[0][e] = src
}

__global__ void k_dinv(float* __restrict__ deg) {
  int i = blockIdx.x * 256 + threadIdx.x;
  if (i < N_NODES) {
    float d = deg[i];
    deg[i] = (d > 0.0f) ? rsqrtf(d) : 0.0f;
  }
}

// ---------------- Acat = [X | LX | h0 | Lh], N x 128 ----------------
__global__ void k_init_acat(const float* __restrict__ X,
                            const float* __restrict__ h0,
                            float* __restrict__ Acat) {
  long idx = (long)blockIdx.x * 256 + threadIdx.x;   // N*128 threads
  int col = (int)(idx & 127);
  long n = idx >> 7;
  float v = 0.0f;
  if (col < 32)                 v = X[n * 32 + col];
  else if (col >= 64 && col < 96) v = h0[n * 32 + (col - 64)];
  Acat[idx] = v;
}

// one wave per edge, lane = feature; coalesced gathers + global_atomic_add_f32
__global__ void k_scatter(const long long* __restrict__ ei,
                          const float* __restrict__ w,
                          const float* __restrict__ dinv,
                          const float* __restrict__ X,
                          const float* __restrict__ h0,
                          float* __restrict__ Acat) {
  long idx = (long)blockIdx.x * 256 + threadIdx.x;
  long e = idx >> 5;
  int f = (int)(idx & 31);
  if (e >= N_EDGES) return;
  long s = (long)ei[e];
  long d = (long)ei[N_EDGES + e];
  float nw = -dinv[s] * w[e] * dinv[d];
  atomicAdd(&Acat[d * 128 + 32 + f], nw * X[s * 32 + f]);   // LX
  atomicAdd(&Acat[d * 128 + 96 + f], nw * h0[s * 32 + f]);  // Lh
}

// ---------------- pack B (weights) into WMMA lane layout ----------------
// Wcat[g][row][n]: row 0..31 -> Wx[g,0], 32..63 -> Wx[g,1], 64..95 -> Wh[g,0], 96..127 -> Wh[g,1]
__device__ __forceinline__ float wcat_elem(const float* __restrict__ Wx,
                                           const float* __restrict__ Wh,
                                           int g, int row, int n) {
  const float* W = (row < 64) ? Wx : Wh;
  int r = row & 63;
  int k = r >> 5;
  int i = r & 31;
  return W[(((g * 2 + k) * 32) + i) * 32 + n];
}

// Bpack[( (g*2+nt)*32 + ks )*32 + lane] = v2f for V_WMMA_F32_16X16X4_F32 B operand:
//   lanes 0-15:  (K=4ks+0, K=4ks+1) for column n0+lane
//   lanes 16-31: (K=4ks+2, K=4ks+3) for column n0+lane-16
__global__ void k_bpack(const float* __restrict__ Wx,
                        const float* __restrict__ Wh,
                        float* __restrict__ Bpack) {
  int idx = blockIdx.x * 256 + threadIdx.x;      // 4*2*32*32 = 8192
  if (idx >= 8192) return;
  int lane = idx & 31;
  int ks = (idx >> 5) & 31;
  int nt = (idx >> 10) & 1;
  int g  = idx >> 11;
  int sel = lane >> 4;
  int n = nt * 16 + (lane & 15);
  int row = ks * 4 + sel * 2;
  v2f v;
  v.x = wcat_elem(Wx, Wh, g, row, n);
  v.y = wcat_elem(Wx, Wh, g, row + 1, n);
  ((v2f*)Bpack)[idx] = v;
}

// ---------------- gate GEMM: Gates[N][128] = Acat(N x 128) @ Wcat(128 x 32) per gate ----------------
// Block = 256 threads = 8 waves; block handles 16 nodes; wave w -> gate w>>1, n-tile w&1.
__global__ __launch_bounds__(256) void k_gemm_gates(const float* __restrict__ Acat,
                                                    const float* __restrict__ Bpack,
                                                    float* __restrict__ Gates) {
  __shared__ float At[16 * 132];                 // pad 128->132 to spread LDS banks
  const int tid = threadIdx.x;
  const long base = (long)blockIdx.x * (16 * 128);

  // Async global->LDS staging of the 16x128 fp32 A-tile (8KB), 16B chunks.
  // 512 chunks, 256 threads x 2. Row padding (132 = 33 float4) keeps 16B alignment.
  {
    const int i0 = tid;
    const int r0 = i0 >> 5, c0 = (i0 & 31) * 4;
    const int i1 = tid + 256;
    const int r1 = i1 >> 5, c1 = (i1 & 31) * 4;
    const float* g0 = Acat + base + (long)r0 * 128 + c0;
    const float* g1 = Acat + base + (long)r1 * 128 + c1;
    unsigned l0 = (unsigned)(size_t)&At[r0 * 132 + c0];   // low 32b of LDS generic ptr = DS offset
    unsigned l1 = (unsigned)(size_t)&At[r1 * 132 + c1];
    asm volatile("global_load_async_to_lds_b128 %0, %1, off"
                 :: "v"(l0), "v"(g0) : "memory");
    asm volatile("global_load_async_to_lds_b128 %0, %1, off"
                 :: "v"(l1), "v"(g1) : "memory");
    asm volatile("s_wait_asynccnt 0x0" ::: "memory");
  }
  __syncthreads();

  const int wave = tid >> 5;
  const int lane = tid & 31;
  const int g  = wave >> 1;
  const int nt = wave & 1;
  const int m  = lane & 15;       // A row (M) for lanes 0-15 / 16-31; also D column (N)
  const int sel = lane >> 4;      // K sub-select

  v8f acc = {};
  const float* arow = At + m * 132 + sel * 2;
  const v2f* bptr = ((const v2f*)Bpack) + ((long)(g * 2 + nt) * 32) * 32 + lane;

  #pragma unroll
  for (int ks = 0; ks < 32; ++ks) {
    v2f a = *(const v2f*)(arow + ks * 4);        // ds_load_b64 (pairs fuse to ds_load_2addr_b64)
    v2f b = bptr[(long)ks * 32];                 // global_load_b64 (L2/L0 hot)
    acc = __builtin_amdgcn_wmma_f32_16x16x4_f32(
        /*neg_a=*/false, a, /*neg_b=*/false, b,
        /*c_mod=*/(short)0, acc, /*reuse_a=*/false, /*reuse_b=*/false);
  }

  // D layout: lanes 0-15 -> N=lane, M=r; lanes 16-31 -> N=lane-16, M=8+r
  const int col = g * 32 + nt * 16 + m;
  const long nodeBase = (long)blockIdx.x * 16 + sel * 8;
  #pragma unroll
  for (int r = 0; r < 8; ++r)
    Gates[(nodeBase + r) * 128 + col] = acc[r];
}

// ---------------- elementwise LSTM + output projection ----------------
__global__ __launch_bounds__(256) void k_lstm_out(const float* __restrict__ Gates,
                                                  const float* __restrict__ c0,
                                                  const float* __restrict__ bx,
                                                  const float* __restrict__ bh,
                                                  const float* __restrict__ b,
                                                  const float* __restrict__ wc,
                                                  const float* __restrict__ W_lin,
                                                  const float* __restrict__ b_lin,
                                                  float* __restrict__ out) {
  __shared__ float sBx[128], sBh[128], sB[128], sWc[96], sWl[32 * 12], sBl[12];
  const int tid = threadIdx.x;
  if (tid < 128) { sBx[tid] = bx[tid]; sBh[tid] = bh[tid]; sB[tid] = b[tid]; }
  if (tid < 96) sWc[tid] = wc[tid];
  if (tid < 12) sBl[tid] = b_lin[tid];
  for (int i = tid; i < 32 * 12; i += 256) sWl[i] = W_lin[i];
  __syncthreads();

  const long n = (long)blockIdx.x * 256 + tid;
  if (n >= N_NODES) return;

  const float* gr = Gates + n * 128;
  float hacc[HORIZON];
  #pragma unroll
  for (int k = 0; k < HORIZON; ++k) hacc[k] = sBl[k];

  float* outH = out + (long)N_NODES * HORIZON + n * 32;
  float* outC = out + (long)N_NODES * HORIZON + (long)N_NODES * 32 + n * 32;

  #pragma unroll 4
  for (int j = 0; j < 32; ++j) {
    float c0j = c0[n * 32 + j];
    float pI = gr[j]       + sBx[j]       + sBh[j]       + sB[j]       + sWc[j] * c0j;
    float pF = gr[32 + j]  + sBx[32 + j]  + sBh[32 + j]  + sB[32 + j]  + sWc[32 + j] * c0j;
    float pT = gr[64 + j]  + sBx[64 + j]  + sBh[64 + j]  + sB[64 + j];
    float I = sigmoidf_(pI);
    float F = sigmoidf_(pF);
    float T = tanhf(pT);
    float C = F * c0j + I * T;
    float pO = gr[96 + j] + sBx[96 + j] + sBh[96 + j] + sB[96 + j] + sWc[64 + j] * C;
    float O = sigmoidf_(pO);
    float H = O * tanhf(C);
    outH[j] = H;
    outC[j] = C;
    float rH = H > 0.0f ? H : 0.0f;
    #pragma unroll
    for (int k = 0; k < HORIZON; ++k) hacc[k] += rH * sWl[j * HORIZON + k];
  }
  float* outh = out + n * HORIZON;
  #pragma unroll
  for (int k = 0; k < HORIZON; ++k) outh[k] = hacc[k];
}

// ---------------- launch ----------------
extern "C" void kernel_launch(void* const* d_in, const int* in_sizes, int n_in,
                              void* d_out, int out_size, void* d_ws, size_t ws_size,
                              hipStream_t stream) {
  const float*     x      = (const float*)d_in[0];      // (N,1,32)
  const long long* ei     = (const long long*)d_in[1];  // (2,E) int64
  const float*     ew     = (const float*)d_in[2];      // (E,)
  const float*     Wx     = (const float*)d_in[3];      // (4,2,32,32)
  const float*     bx     = (const float*)d_in[4];      // (4,32)
  const float*     Wh     = (const float*)d_in[5];      // (4,2,32,32)
  const float*     bh     = (const float*)d_in[6];      // (4,32)
  const float*     wc     = (const float*)d_in[7];      // (3,32)
  const float*     b      = (const float*)d_in[8];      // (4,32)
  const float*     W_lin  = (const float*)d_in[9];      // (32,12)
  const float*     b_lin  = (const float*)d_in[10];     // (12,)
  const float*     h0     = (const float*)d_in[11];     // (N,32)
  const float*     c0     = (const float*)d_in[12];     // (N,32)
  float* out = (float*)d_out;

  // workspace layout (floats): deg/dinv | Acat N*128 | Gates N*128 | Bpack 4*2*32*32*2
  float* ws    = (float*)d_ws;
  float* deg   = ws;                          // 50000 (reused as dinv)
  float* Acat  = ws + 50176;                  // 6,400,000
  float* Gates = Acat + (long)N_NODES * 128;  // 6,400,000
  float* Bpack = Gates + (long)N_NODES * 128; // 16,384

  const int T = 256;
  k_zero_deg <<<(N_NODES + T - 1) / T, T, 0, stream>>>(deg);
  k_deg_accum<<<(N_EDGES + T - 1) / T, T, 0, stream>>>(ei, ew, deg);
  k_dinv     <<<(N_NODES + T - 1) / T, T, 0, stream>>>(deg);
  k_init_acat<<<(N_NODES * 128) / T, T, 0, stream>>>(x, h0, Acat);
  k_bpack    <<<8192 / T, T, 0, stream>>>(Wx, Wh, Bpack);
  k_scatter  <<<(int)(((long)N_EDGES * 32) / T), T, 0, stream>>>(ei, ew, deg, x, h0, Acat);
  k_gemm_gates<<<N_NODES / 16, T, 0, stream>>>(Acat, Bpack, Gates);
  k_lstm_out <<<(N_NODES + T - 1) / T, T, 0, stream>>>(Gates, c0, bx, bh, b, wc, W_lin, b_lin, out);
}